// Model_83519934038718
// MI455X (gfx1250) — compile-verified
//
#include <hip/hip_runtime.h>

typedef float v2f __attribute__((ext_vector_type(2)));
typedef float v8f __attribute__((ext_vector_type(8)));

#define NN 4096
#define PP 256
#define MM 64
#define OO 40
#define EE 131072
#define KAPPA_C 0.99f
#define TOL_C 3e-6f
#define MITR_C 300

// ---------------------------------------------------------------------------
__global__ __launch_bounds__(256) void zero4_kernel(float4* __restrict__ p, int n4) {
    int i = blockIdx.x * blockDim.x + threadIdx.x;
    if (i < n4) p[i] = make_float4(0.f, 0.f, 0.f, 0.f);
}

// scatter-add edges into dense A (row = src, col = dst)
__global__ __launch_bounds__(256) void scatter_kernel(const int* __restrict__ ei,
                                                      const float* __restrict__ ew,
                                                      float* __restrict__ A) {
    int e = blockIdx.x * blockDim.x + threadIdx.x;
    if (e >= EE) return;
    int i = ei[e];
    int j = ei[EE + e];
    atomicAdd(A + (size_t)i * NN + j, ew[e]);
}

// per-row projection of W (64x64) onto the l1-ball of radius KAPPA
__global__ void proj_kernel(const float* __restrict__ W, float* __restrict__ Wp) {
    int r = threadIdx.x;
    if (r >= MM) return;
    float a[MM];
    float s = 0.f;
    for (int j = 0; j < MM; ++j) { a[j] = fabsf(W[r * MM + j]); s += a[j]; }
    if (s <= KAPPA_C) {
        for (int j = 0; j < MM; ++j) Wp[r * MM + j] = W[r * MM + j];
        return;
    }
    for (int i = 1; i < MM; ++i) {  // insertion sort, descending
        float key = a[i];
        int j = i - 1;
        while (j >= 0 && a[j] < key) { a[j + 1] = a[j]; --j; }
        a[j + 1] = key;
    }
    float css = 0.f;
    int cnt = 0;
    for (int k = 0; k < MM; ++k) {
        css += a[k];
        if (a[k] * (float)(k + 1) > css - KAPPA_C) ++cnt;
    }
    if (cnt < 1) cnt = 1;
    float c2 = 0.f;
    for (int k = 0; k < cnt; ++k) c2 += a[k];
    float theta = (c2 - KAPPA_C) / (float)cnt;
    for (int j = 0; j < MM; ++j) {
        float w = W[r * MM + j];
        float m = fmaxf(fabsf(w) - theta, 0.f);
        Wp[r * MM + j] = (w > 0.f) ? m : ((w < 0.f) ? -m : 0.f);
    }
}

__global__ void init_flags_kernel(unsigned* flags) { flags[0] = 0u; flags[1] = 0u; }

// read err; if converged set done; reset err for next iteration
__global__ void ctrl_kernel(unsigned* flags) {
    if (!flags[1] && __uint_as_float(flags[0]) < TOL_C) flags[1] = 1u;
    flags[0] = 0u;
}

// X = Xnew unless converged (torch breaks BEFORE assigning)
__global__ __launch_bounds__(256) void copy_if_kernel(const float4* __restrict__ src,
                                                      float4* __restrict__ dst,
                                                      const unsigned* __restrict__ done) {
    if (*done) return;
    int i = blockIdx.x * blockDim.x + threadIdx.x;  // exactly 64*4096/4 threads
    dst[i] = src[i];
}

// ---------------------------------------------------------------------------
// OUT(64 x 4096) = [relu]( IN(64 x K) @ B(K x 4096) [+ bias] )
// TRANSB: B stored as (4096 x ldb), element B[k][n] = MB[n*ldb + k].
// DO_ERR: atomicMax(errbits, max|OUT - Xprev|) (nonneg float bits order as uint).
// CHECK_DONE: early-exit when *done (one scalar load per wave).
// Each wave computes a 32x64 slab: 8 v_wmma_f32_16x16x4_f32 per K-step.
// All addresses = uniform(k0) + lane-constant offset -> GVS saddr loads,
// scalar-only address updates in the steady-state loop.
// Launch: 16 blocks x 256 threads = 128 waves (2 row slabs x 64 col groups).
template <int TRANSB, int HAS_BIAS, int DO_RELU, int DO_ERR, int CHECK_DONE>
__global__ __launch_bounds__(256) void gemm64_wmma(
    const float* __restrict__ IN, int lda,
    const float* __restrict__ MB, int ldb,
    float* __restrict__ OUT,
    const float* __restrict__ bias,
    const float* __restrict__ Xprev,
    unsigned* __restrict__ errbits,
    const unsigned* __restrict__ done,
    int K) {
    if (CHECK_DONE && *done) return;

    const int lane = threadIdx.x & 31;
    const int wave = (blockIdx.x * (int)blockDim.x + threadIdx.x) >> 5;  // 0..127
    const int mt = wave >> 6;                                            // 0..1
    const int ng = wave & 63;                                            // 0..63
    const int m0 = mt * 32;
    const int n0 = ng * 64;
    const int l15 = lane & 15;
    const int lhi = lane >> 4;  // 0 or 1

    v8f acc0[4], acc1[4];
#pragma unroll
    for (int t = 0; t < 4; ++t)
#pragma unroll
        for (int r = 0; r < 8; ++r) { acc0[t][r] = 0.f; acc1[t][r] = 0.f; }

    // lane-constant offsets (in floats)
    // A-operand 16x4: lanes 0-15 -> K={k0,k0+1}, lanes 16-31 -> K={k0+2,k0+3}
    const unsigned aOff = (unsigned)((m0 + l15) * lda + 2 * lhi);  // row m0+l15
    const unsigned aOff1 = aOff + (unsigned)(16 * lda);            // row m0+16+l15
    // B-operand 4x16: VGPR0 rows {k0,k0+2}, VGPR1 rows {k0+1,k0+3}
    const unsigned bOff = TRANSB ? (unsigned)((n0 + l15) * ldb + 2 * lhi)
                                 : (unsigned)(2 * lhi * ldb + n0 + l15);

#pragma unroll 2
    for (int k0 = 0; k0 < K; k0 += 4) {
        const float* aK = IN + (size_t)k0;         // uniform part
        const float* bK = MB + (size_t)k0 * (TRANSB ? 1 : ldb);
        v2f a0 = *(const v2f*)(aK + aOff);
        v2f a1 = *(const v2f*)(aK + aOff1);
#pragma unroll
        for (int t = 0; t < 4; ++t) {
            v2f b;
            if (TRANSB) {
                b = *(const v2f*)(bK + bOff + (unsigned)(t * 16 * ldb));
            } else {
                b[0] = bK[bOff + (unsigned)(t * 16)];
                b[1] = bK[bOff + (unsigned)(t * 16) + (unsigned)ldb];
            }
            acc0[t] = __builtin_amdgcn_wmma_f32_16x16x4_f32(
                false, a0, false, b, (short)0, acc0[t], false, false);
            acc1[t] = __builtin_amdgcn_wmma_f32_16x16x4_f32(
                false, a1, false, b, (short)0, acc1[t], false, false);
        }
    }

    // C/D layout: VGPR r -> row (r + 8*lhi), lanes 0-15 -> N = tile + lane
    float emax = 0.f;
#pragma unroll
    for (int t = 0; t < 4; ++t) {
        const int col = n0 + t * 16 + l15;
#pragma unroll
        for (int r = 0; r < 8; ++r) {
            int row0 = m0 + lhi * 8 + r;
            int row1 = row0 + 16;
            size_t i0 = (size_t)row0 * NN + col;
            size_t i1 = (size_t)row1 * NN + col;
            float v0 = acc0[t][r];
            float v1 = acc1[t][r];
            if (HAS_BIAS) { v0 += bias[i0]; v1 += bias[i1]; }
            if (DO_RELU) { v0 = fmaxf(v0, 0.f); v1 = fmaxf(v1, 0.f); }
            if (DO_ERR) {
                emax = fmaxf(emax, fabsf(v0 - Xprev[i0]));
                emax = fmaxf(emax, fabsf(v1 - Xprev[i1]));
            }
            OUT[i0] = v0;
            OUT[i1] = v1;
        }
    }
    if (DO_ERR) atomicMax(errbits, __float_as_uint(emax));
}

// ---------------------------------------------------------------------------
// out(n, o) = dot(normalize(Xf[:, n]), V_W[o, :])
__global__ __launch_bounds__(256) void out_kernel(const float* __restrict__ Xf,
                                                  const float* __restrict__ VW,
                                                  float* __restrict__ out) {
    __shared__ float vsh[OO * MM];
    for (int i = threadIdx.x; i < OO * MM; i += blockDim.x) vsh[i] = VW[i];
    __syncthreads();
    int n = blockIdx.x * blockDim.x + threadIdx.x;  // 4096 threads
    float h[MM];
    float ss = 0.f;
#pragma unroll 8
    for (int m = 0; m < MM; ++m) {
        float v = Xf[(size_t)m * NN + n];
        h[m] = v;
        ss += v * v;
    }
    float inv = 1.0f / fmaxf(sqrtf(ss), 1e-12f);
    for (int o = 0; o < OO; ++o) {
        float d = 0.f;
#pragma unroll 8
        for (int m = 0; m < MM; ++m) d += h[m] * vsh[o * MM + m];
        out[(size_t)n * OO + o] = d * inv;
    }
}

// ---------------------------------------------------------------------------
extern "C" void kernel_launch(void* const* d_in, const int* in_sizes, int n_in,
                              void* d_out, int out_size, void* d_ws, size_t ws_size,
                              hipStream_t stream) {
    (void)in_sizes; (void)n_in; (void)out_size; (void)ws_size;

    const float* x   = (const float*)d_in[0];  // (4096, 256)
    const int*   ei  = (const int*)d_in[1];    // (2, 131072)
    const float* ew  = (const float*)d_in[2];  // (131072,)
    const float* W   = (const float*)d_in[3];  // (64, 64)
    const float* Om1 = (const float*)d_in[4];  // (64, 256)
    // d_in[5] = Omega_2 (unused by the reference forward)
    const float* VW  = (const float*)d_in[6];  // (40, 64)
    const float* X0  = (const float*)d_in[7];  // (64, 4096)
    float* outp = (float*)d_out;               // (4096, 40)

    // workspace carve (256B aligned)
    char* wsp = (char*)d_ws;
    auto carve = [&](size_t bytes) -> void* {
        void* p = (void*)wsp;
        wsp += (bytes + 255) & ~(size_t)255;
        return p;
    };
    float* A   = (float*)carve((size_t)NN * NN * sizeof(float));  // 64 MB (L2-resident)
    float* T   = (float*)carve((size_t)MM * NN * sizeof(float));
    float* X   = (float*)carve((size_t)MM * NN * sizeof(float));
    float* Xn  = (float*)carve((size_t)MM * NN * sizeof(float));
    float* Bb  = (float*)carve((size_t)MM * NN * sizeof(float));
    float* C1  = (float*)carve((size_t)MM * NN * sizeof(float));
    float* Wp  = (float*)carve((size_t)MM * MM * sizeof(float));
    unsigned* flags = (unsigned*)carve(256);  // [0]=err bits, [1]=done

    const int GEMM_BLOCKS = 16;  // 128 waves
    const int THREADS = 256;

    // ---- setup ----
    hipMemcpyAsync(X, X0, (size_t)MM * NN * sizeof(float),
                   hipMemcpyDeviceToDevice, stream);
    zero4_kernel<<<(NN * NN / 4) / THREADS, THREADS, 0, stream>>>(
        (float4*)A, NN * NN / 4);
    scatter_kernel<<<EE / THREADS, THREADS, 0, stream>>>(ei, ew, A);
    proj_kernel<<<1, 64, 0, stream>>>(W, Wp);
    init_flags_kernel<<<1, 1, 0, stream>>>(flags);

    // C1 = Omega_1 @ x^T   (K = 256, B transposed: U[k][n] = x[n*256+k])
    gemm64_wmma<1, 0, 0, 0, 0><<<GEMM_BLOCKS, THREADS, 0, stream>>>(
        Om1, PP, x, PP, C1, nullptr, nullptr, nullptr, nullptr, PP);
    // B = C1 @ A
    gemm64_wmma<0, 0, 0, 0, 0><<<GEMM_BLOCKS, THREADS, 0, stream>>>(
        C1, NN, A, NN, Bb, nullptr, nullptr, nullptr, nullptr, NN);

    // ---- fixed-point loop (device-side early exit via done flag) ----
    for (int it = 0; it < MITR_C; ++it) {
        // T = Wp @ X  (K = 64)
        gemm64_wmma<0, 0, 0, 0, 1><<<GEMM_BLOCKS, THREADS, 0, stream>>>(
            Wp, MM, X, NN, T, nullptr, nullptr, nullptr, flags + 1, MM);
        // Xn = relu(T @ A + B), err = max|Xn - X|
        gemm64_wmma<0, 1, 1, 1, 1><<<GEMM_BLOCKS, THREADS, 0, stream>>>(
            T, NN, A, NN, Xn, Bb, X, flags + 0, flags + 1, NN);
        ctrl_kernel<<<1, 1, 0, stream>>>(flags);
        copy_if_kernel<<<(MM * NN / 4) / THREADS, THREADS, 0, stream>>>(
            (const float4*)Xn, (float4*)X, flags + 1);
    }

    // ---- final Xf = f(X) (always executed) ----
    gemm64_wmma<0, 0, 0, 0, 0><<<GEMM_BLOCKS, THREADS, 0, stream>>>(
        Wp, MM, X, NN, T, nullptr, nullptr, nullptr, nullptr, MM);
    gemm64_wmma<0, 1, 1, 0, 0><<<GEMM_BLOCKS, THREADS, 0, stream>>>(
        T, NN, A, NN, Xn, Bb, nullptr, nullptr, nullptr, NN);

    // ---- normalize + linear epilogue ----
    out_kernel<<<NN / THREADS, THREADS, 0, stream>>>(Xn, VW, outp);
}